// AverageSeqEncoder_2671469658392
// MI455X (gfx1250) — compile-verified
//
#include <hip/hip_runtime.h>

typedef float v2f __attribute__((ext_vector_type(2)));
typedef float v8f __attribute__((ext_vector_type(8)));

// Problem shape (fixed by the reference): inputs (8, 4096, 1024) f32.
constexpr int B_  = 8;
constexpr int T_  = 4096;
constexpr int C_  = 1024;
constexpr int NCH = 16;          // chunks along T
constexpr int TC  = T_ / NCH;    // 256 timesteps per chunk
constexpr int TILES = TC / 16;   // 16 WMMA tiles (16 timesteps each) per chunk

// ---------------------------------------------------------------------------
// Kernel 1: per-(b, chunk, c) sum of the chunk. Fully coalesced streaming
// read of the input (also primes the 192MB L2 for kernel 2's re-read).
// ---------------------------------------------------------------------------
__global__ __launch_bounds__(256) void k_chunk_sums(const float* __restrict__ in,
                                                    float* __restrict__ ws) {
    const int idx   = blockIdx.x * 256 + threadIdx.x;  // 0 .. 131071
    const int c     = idx & (C_ - 1);
    const int chunk = (idx >> 10) & (NCH - 1);
    const int b     = idx >> 14;

    const float* p = in + ((size_t)(b * T_ + chunk * TC) * C_ + c);
    float s0 = 0.f, s1 = 0.f, s2 = 0.f, s3 = 0.f;
    for (int t = 0; t < TC; t += 4) {
        s0 += p[(size_t)(t + 0) * C_];
        s1 += p[(size_t)(t + 1) * C_];
        s2 += p[(size_t)(t + 2) * C_];
        s3 += p[(size_t)(t + 3) * C_];
    }
    ws[(b * NCH + chunk) * C_ + c] = (s0 + s1) + (s2 + s3);
}

// ---------------------------------------------------------------------------
// Kernel 2: one wave32 per (b, 16-channel tile, chunk).
// Inclusive cumsum of each 16(t) x 16(c) tile as D = L * X + carry via
// 4x V_WMMA_F32_16X16X4_F32 (L = lower-triangular ones, held in const VGPRs).
// Carry chains across tiles via a wave shuffle of D's row 15.
// ---------------------------------------------------------------------------
__global__ __launch_bounds__(256) void k_scan_wmma(const float* __restrict__ in,
                                                   const float* __restrict__ ws,
                                                   float* __restrict__ out) {
    const int lane  = threadIdx.x & 31;
    const int wave  = threadIdx.x >> 5;
    const int w     = blockIdx.x * 8 + wave;     // 0 .. 8191
    const int chunk = w & (NCH - 1);
    const int ctile = (w >> 4) & 63;
    const int b     = w >> 10;

    const int n     = lane & 15;   // matrix column == channel within tile
    const int khalf = lane >> 4;   // 0: K/M rows 0..1 / M=vgpr ; 1: K rows 2..3 / M=vgpr+8
    const int cidx  = ctile * 16 + n;

    // A panels of L (16x4 each, lower-triangular ones): A[k][m][j] = (4k+j <= m).
    // A layout: lane 0-15 -> M=lane, VGPR j holds K = 2*khalf + j.
    v2f A[4];
#pragma unroll
    for (int k = 0; k < 4; ++k) {
        A[k].x = ((4 * k + 2 * khalf + 0) <= n) ? 1.0f : 0.0f;
        A[k].y = ((4 * k + 2 * khalf + 1) <= n) ? 1.0f : 0.0f;
    }

    // Initial carry for this chunk: prefix sum of earlier chunk sums (per channel).
    float carry = 0.0f;
    {
        const float* wsp = ws + (size_t)b * NCH * C_ + cidx;
        for (int k = 0; k < chunk; ++k) carry += wsp[(size_t)k * C_];
    }

    const size_t baseBT = (size_t)b * T_;

    for (int tile = 0; tile < TILES; ++tile) {
        const int tbase = chunk * TC + tile * 16;

        // C init: every row gets this lane's per-channel carry.
        v8f acc;
#pragma unroll
        for (int r = 0; r < 8; ++r) acc[r] = carry;

        // D = sum_k A[k] (16x4) * X[k] (4x16) + C
        // B layout: lanes 0-15 -> N=lane, VGPR j holds K = 2*khalf + j.
#pragma unroll
        for (int kb = 0; kb < 4; ++kb) {
            const int t0   = tbase + 4 * kb + 2 * khalf;
            const float* p = in + ((baseBT + (size_t)t0) * C_ + cidx);
            v2f Bm;
            Bm.x = p[0];
            Bm.y = p[C_];
            acc = __builtin_amdgcn_wmma_f32_16x16x4_f32(
                /*neg_a=*/false, A[kb], /*neg_b=*/false, Bm,
                /*c_mod=*/(short)0, acc, /*reuse_a=*/false, /*reuse_b=*/false);
        }

        // Carry for next tile: D[M=15][N=n] lives in VGPR7, lane 16+n.
        carry = __shfl(acc[7], 16 + n, 32);

        // Scale by 1/(t+1) (v_rcp_f32, ~1 ulp) and store.
        float* q = out + ((baseBT + (size_t)(tbase + khalf * 8)) * C_ + cidx);
#pragma unroll
        for (int r = 0; r < 8; ++r) {
            const int t = tbase + khalf * 8 + r;
            const float inv = __builtin_amdgcn_rcpf((float)(t + 1));
            q[(size_t)r * C_] = acc[r] * inv;
        }
    }
}

// ---------------------------------------------------------------------------
extern "C" void kernel_launch(void* const* d_in, const int* in_sizes, int n_in,
                              void* d_out, int out_size, void* d_ws, size_t ws_size,
                              hipStream_t stream) {
    const float* in = (const float*)d_in[0];
    float* out      = (float*)d_out;
    float* ws       = (float*)d_ws;   // needs B_*NCH*C_*4 = 512 KB

    // Kernel 1: 8*16*1024 = 131072 threads.
    k_chunk_sums<<<(B_ * NCH * C_) / 256, 256, 0, stream>>>(in, ws);

    // Kernel 2: 8192 waves = 1024 blocks of 256 threads (8 waves each).
    k_scan_wmma<<<(B_ * 64 * NCH) / 8, 256, 0, stream>>>(in, ws, out);
}